// MMCLHead_47914655154323
// MI455X (gfx1250) — compile-verified
//
#include <hip/hip_runtime.h>
#include <cstdint>

// ---------------------------------------------------------------------------
// MMCL head loss for MI455X (gfx1250).
//   kernel 1: one workgroup per row.  TDM-DMAs the whole 256KB logit row into
//             LDS, streams labels from HBM once, then does an exact 3-level
//             radix top-k select entirely out of LDS using packed u64
//             {count:16 | fixedpoint sum(1+x)^2 : 48} histograms (ds_add_u64,
//             deterministic).  Writes per-row loss to d_ws.
//   kernel 2: 1024 -> 1 mean via v_wmma_f32_16x16x4_f32 (A = ones), layout-
//             agnostic because sum over all D slots = 16 * sum(B entries).
// HBM traffic: 256MB (TDM) + 256MB (labels) = 512MB -> ~22us at 23.3 TB/s.
// ---------------------------------------------------------------------------

#define MDIM    65536
#define NROWS   1024
#define TPB     512
#define NBINS   4096
#define NCHUNK  512                                   // NBINS / 8
#define SMEM_BYTES (MDIM*4 + NBINS*8 + NCHUNK*8)      // 262144+32768+4096=299008

#define SCALE_F   268435456.0f                        // 2^28
#define INV_SCALE 3.7252902984619141e-09f             // 2^-28
#define M48       0xFFFFFFFFFFFFULL
#define ONE48     (1ULL << 48)

typedef unsigned int u32x4 __attribute__((ext_vector_type(4)));
typedef int          i32x8 __attribute__((ext_vector_type(8)));
typedef int          i32x4 __attribute__((ext_vector_type(4)));
typedef float        v2f   __attribute__((ext_vector_type(2)));
typedef float        v8f   __attribute__((ext_vector_type(8)));

// monotonic float-bits -> uint key (ascending float => ascending key)
__device__ __forceinline__ uint32_t val_to_key(uint32_t u) {
  return (u & 0x80000000u) ? ~u : (u | 0x80000000u);
}
__device__ __forceinline__ float key_to_val(uint32_t key) {
  uint32_t u = (key & 0x80000000u) ? (key & 0x7FFFFFFFu) : ~key;
  return __uint_as_float(u);
}
// packed histogram entry: {count=1 in [63:48]} | {(1+x)^2 * 2^28 in [47:0]}
__device__ __forceinline__ unsigned long long pack_term(float x) {
  float t = 1.0f + x;
  return ONE48 | (unsigned long long)(t * t * SCALE_F);
}

__global__ __launch_bounds__(TPB)
void mmcl_row_kernel(const float* __restrict__ logits,
                     const int*  __restrict__ labels,
                     float* __restrict__ row_loss) {
  extern __shared__ unsigned char smem[];
  uint32_t*           keys  = (uint32_t*)smem;                           // 65536 u32
  unsigned long long* hist  = (unsigned long long*)(smem + MDIM * 4);    // 4096 u64
  unsigned long long* chunk = (unsigned long long*)(smem + MDIM * 4 + NBINS * 8); // 512 u64
  __shared__ unsigned long long s_pos;   // packed {pos_cnt | pos_sum*2^28}
  __shared__ float        s_sum;         // accumulated top-k sum (thread0 only)
  __shared__ unsigned int s_b, s_rem, s_pre, s_k;

  const int tid = threadIdx.x;
  const int row = blockIdx.x;

  // ---- stage the whole row in LDS with one TDM descriptor (wave 0 only) ----
  if (tid < 32) {
    uint64_t ga = (uint64_t)(uintptr_t)(logits + (size_t)row * MDIM);
    uint32_t lds_off = (uint32_t)(uintptr_t)keys;  // flat addr low 32b == LDS byte offset
    u32x4 g0;
    g0[0] = 1u;                                        // count = 1 (valid D#)
    g0[1] = lds_off;                                   // lds_addr (bytes)
    g0[2] = (uint32_t)ga;                              // global_addr[31:0]
    g0[3] = (uint32_t)((ga >> 32) & 0x01FFFFFFu) | (2u << 30); // addr[56:32] | type=2
    i32x8 g1;
    g1[0] = 0x00020000;   // workgroup_mask=0, data_size=2 (4 bytes)
    g1[1] = 0x20000000;   // tensor_dim0 = 8192 (low16 in bits[31:16])
    g1[2] = 0x00080000;   // tensor_dim0 hi = 0, tensor_dim1 = 8
    g1[3] = 0x20000000;   // tensor_dim1 hi = 0, tile_dim0 = 8192
    g1[4] = 0x00000008;   // tile_dim1 = 8, tile_dim2 = 0
    g1[5] = 8192;         // tensor_dim0_stride = 8192
    g1[6] = 0;
    g1[7] = 0;
    i32x4 gz; gz[0] = 0; gz[1] = 0; gz[2] = 0; gz[3] = 0;
#if __has_include(<hip/amd_detail/amd_gfx1250_TDM.h>)
    i32x8 gz8; for (int i = 0; i < 8; ++i) gz8[i] = 0;
    __builtin_amdgcn_tensor_load_to_lds(g0, g1, gz, gz, gz8, 0);
#else
    __builtin_amdgcn_tensor_load_to_lds(g0, g1, gz, gz, 0);
#endif
  }
  // clear histograms / scalars while the DMA is in flight
  for (int i = tid; i < NBINS; i += TPB) hist[i] = 0ULL;
  if (tid == 0) { s_pos = 0ULL; s_sum = 0.0f; }
  if (tid < 32) __builtin_amdgcn_s_wait_tensorcnt(0);
  __syncthreads();

  // ---- pass 1: labels from HBM (B128), logits from LDS; build L1 histogram ----
  {
    const int4* lab4 = (const int4*)(labels + (size_t)row * MDIM);
    float lpos = 0.0f; unsigned int lcnt = 0;
    for (int it = 0; it < MDIM / 4 / TPB; ++it) {        // 32 iterations
      int c4 = tid + it * TPB;
      int4 lv = lab4[c4];
      int lab[4] = { lv.x, lv.y, lv.z, lv.w };
      int base = c4 * 4;
#pragma unroll
      for (int j = 0; j < 4; ++j) {
        uint32_t u = keys[base + j];                     // raw float bits (via TDM)
        float x = __uint_as_float(u);
        if (lab[j]) {
          float d = 1.0f - x;
          lpos += d * d; lcnt++;
          keys[base + j] = 0u;                           // sentinel: smallest key
        } else {
          uint32_t key = val_to_key(u);
          keys[base + j] = key;
          atomicAdd(&hist[key >> 20], pack_term(x));     // ds_add_u64
        }
      }
    }
    if (lcnt)
      atomicAdd(&s_pos, ((unsigned long long)lcnt << 48) |
                        (unsigned long long)(lpos * SCALE_F));
  }
  __syncthreads();

  // ---- k = floor(0.01 * neg_cnt); chunk pre-reduction for L1 select ----
  if (tid == 0) {
    unsigned int pos_cnt = (unsigned int)(s_pos >> 48);
    unsigned int neg_cnt = MDIM - pos_cnt;
    unsigned int k = (unsigned int)floorf(0.01f * (float)neg_cnt);
    s_k = k; s_rem = k;
  }
  {
    unsigned long long t = 0ULL;
#pragma unroll
    for (int j = 0; j < 8; ++j) t += hist[tid * 8 + j];
    chunk[tid] = t;
  }
  __syncthreads();

  // ---- L1 select (top 12 key bits) ----
  if (tid == 0) {
    unsigned int r = s_rem;
    unsigned long long cum = 0ULL;
    int c = NCHUNK - 1;
    for (; c > 0; --c) {
      unsigned long long t = chunk[c];
      if ((unsigned int)((cum + t) >> 48) >= r) break;
      cum += t;
    }
    int b = c * 8;
    for (int j = 7; j >= 0; --j) {
      unsigned long long t = hist[c * 8 + j];
      if (j == 0 || (unsigned int)((cum + t) >> 48) >= r) { b = c * 8 + j; break; }
      cum += t;
    }
    s_b   = (unsigned int)b;
    s_rem = r - (unsigned int)(cum >> 48);
    s_sum += (float)(cum & M48) * INV_SCALE;             // everything above bin b
  }
  __syncthreads();
  const unsigned int b1 = s_b;

  // ---- pass 2 (LDS only): mid 12 key bits within tie bin b1 ----
  for (int i = tid; i < NBINS; i += TPB) hist[i] = 0ULL;
  __syncthreads();
  for (int i = tid; i < MDIM; i += TPB) {
    uint32_t key = keys[i];
    if ((key >> 20) == b1)
      atomicAdd(&hist[(key >> 8) & 0xFFFu], pack_term(key_to_val(key)));
  }
  __syncthreads();
  {
    unsigned long long t = 0ULL;
#pragma unroll
    for (int j = 0; j < 8; ++j) t += hist[tid * 8 + j];
    chunk[tid] = t;
  }
  __syncthreads();
  if (tid == 0) {
    unsigned int r = s_rem;
    unsigned long long cum = 0ULL;
    int c = NCHUNK - 1;
    for (; c > 0; --c) {
      unsigned long long t = chunk[c];
      if ((unsigned int)((cum + t) >> 48) >= r) break;
      cum += t;
    }
    int b = c * 8;
    for (int j = 7; j >= 0; --j) {
      unsigned long long t = hist[c * 8 + j];
      if (j == 0 || (unsigned int)((cum + t) >> 48) >= r) { b = c * 8 + j; break; }
      cum += t;
    }
    s_pre = (b1 << 12) | (unsigned int)b;                // top-24 prefix of threshold
    s_rem = r - (unsigned int)(cum >> 48);
    s_sum += (float)(cum & M48) * INV_SCALE;
  }
  __syncthreads();
  const unsigned int pre24 = s_pre;

  // ---- pass 3 (LDS only): final 8 key bits within tie prefix ----
  for (int i = tid; i < 256; i += TPB) hist[i] = 0ULL;
  __syncthreads();
  for (int i = tid; i < MDIM; i += TPB) {
    uint32_t key = keys[i];
    if ((key >> 8) == pre24)
      atomicAdd(&hist[key & 0xFFu], pack_term(key_to_val(key)));
  }
  __syncthreads();

  // ---- L3 select + finalize row loss ----
  if (tid == 0) {
    unsigned int r = s_rem;
    unsigned long long cum = 0ULL;
    int b = 0;
    for (int j = 255; j >= 0; --j) {
      unsigned long long t = hist[j];
      if (j == 0 || (unsigned int)((cum + t) >> 48) >= r) { b = j; break; }
      cum += t;
    }
    unsigned int rem3 = r - (unsigned int)(cum >> 48);   // taken from exact-tie key
    float sum = s_sum + (float)(cum & M48) * INV_SCALE;
    uint32_t T = (pre24 << 8) | (unsigned int)b;         // full 32-bit threshold key
    float tt = 1.0f + key_to_val(T);
    sum += (float)rem3 * tt * tt;                        // ties: identical values

    unsigned int pos_cnt = (unsigned int)(s_pos >> 48);
    float pos_sum = (float)(s_pos & M48) * INV_SCALE;
    float loss = 5.0f * pos_sum / (float)pos_cnt + sum / (float)s_k;
    row_loss[row] = loss;
  }
}

// ---- kernel 2: mean of 1024 row losses via the matrix pipe ----------------
// A = ones(16x4) so D = A*B + C has every row equal to the column sums of B;
// sum over all 256 D slots = 16 * sum(B entries)  (independent of VGPR layout).
__global__ __launch_bounds__(32)
void mmcl_reduce_kernel(const float* __restrict__ row_loss,
                        float* __restrict__ out) {
  const int lane = threadIdx.x;                          // full wave32, EXEC all 1s
  v8f acc; 
#pragma unroll
  for (int i = 0; i < 8; ++i) acc[i] = 0.0f;
  v2f a; a[0] = 1.0f; a[1] = 1.0f;
  for (int t = 0; t < NROWS / 64; ++t) {                 // 16 iterations x 64 values
    v2f b;
    b[0] = row_loss[t * 64 + lane];
    b[1] = row_loss[t * 64 + 32 + lane];
    acc = __builtin_amdgcn_wmma_f32_16x16x4_f32(false, a, false, b,
                                                (short)0, acc, false, false);
  }
  float p = 0.0f;
#pragma unroll
  for (int i = 0; i < 8; ++i) p += acc[i];
#pragma unroll
  for (int off = 16; off > 0; off >>= 1) p += __shfl_down(p, off, 32);
  if (lane == 0) out[0] = p * (1.0f / 16.0f) * (1.0f / (float)NROWS);
}

extern "C" void kernel_launch(void* const* d_in, const int* in_sizes, int n_in,
                              void* d_out, int out_size, void* d_ws, size_t ws_size,
                              hipStream_t stream) {
  (void)in_sizes; (void)n_in; (void)out_size; (void)ws_size;
  const float* logits = (const float*)d_in[0];
  const int*   labels = (const int*)d_in[1];
  float* rowloss = (float*)d_ws;           // 1024 floats of scratch
  float* out     = (float*)d_out;

  (void)hipFuncSetAttribute((const void*)mmcl_row_kernel,
                            hipFuncAttributeMaxDynamicSharedMemorySize,
                            SMEM_BYTES);
  mmcl_row_kernel<<<NROWS, TPB, SMEM_BYTES, stream>>>(logits, labels, rowloss);
  mmcl_reduce_kernel<<<1, 32, 0, stream>>>(rowloss, out);
}